// Test_Model_26388279066905
// MI455X (gfx1250) — compile-verified
//
#include <hip/hip_runtime.h>

typedef __attribute__((ext_vector_type(2))) float v2f;
typedef __attribute__((ext_vector_type(8))) float v8f;

#define NL 7        // output l = 0..6
#define LMAXP1 4    // input parts l = 0..3
#define TAU 64
#define BATCH 64

struct Combo {
  int l, l1, l2, fi, r, K;   // output l, fragment (l1,l2), frag index, r=m+l, #pairs
  int p[8];                  // m1+l1 for each pair
  int q[8];                  // m2+l2 for each pair
  float c[8];                // CG coefficient per pair
  int pad[2];
};
static_assert(sizeof(Combo) == 128, "combo must be 128B");

__device__ __constant__ int d_NFRAG[NL] = {4, 9, 11, 10, 6, 3, 1};
__device__ __constant__ unsigned long long d_OUTBASE[NL] = {
    0ull, 2097152ull, 16252928ull, 45088768ull,
    81788928ull, 110100480ull, 127401984ull};

__device__ double dfact(int n) {
  double r = 1.0;
  for (int i = 2; i <= n; ++i) r *= (double)i;
  return r;
}

// Racah formula, matches reference _cg()
__device__ double cg_coef(int j1, int m1, int j2, int m2, int j, int m) {
  if (m1 + m2 != m) return 0.0;
  double pre = sqrt((double)(2 * j + 1) * dfact(j + j1 - j2) * dfact(j - j1 + j2) *
                    dfact(j1 + j2 - j) / dfact(j1 + j2 + j + 1));
  pre *= sqrt(dfact(j + m) * dfact(j - m) * dfact(j1 - m1) * dfact(j1 + m1) *
              dfact(j2 - m2) * dfact(j2 + m2));
  double s = 0.0;
  for (int k = 0; k <= j1 + j2 - j; ++k) {
    int d2 = j1 + j2 - j - k, d3 = j1 - m1 - k, d4 = j2 + m2 - k;
    int d5 = j - j2 + m1 + k, d6 = j - j1 - m2 + k;
    if (d2 < 0 || d3 < 0 || d4 < 0 || d5 < 0 || d6 < 0) continue;
    double den = dfact(k) * dfact(d2) * dfact(d3) * dfact(d4) * dfact(d5) * dfact(d6);
    s += ((k & 1) ? -1.0 : 1.0) / den;
  }
  return pre * s;
}

// One thread per combo: enumerate (l, l1, l2, r) in reference fragment order,
// then list all (m1,m2) pairs with m1+m2=m and their CG coefficients.
__global__ void cg_setup(Combo* __restrict__ combos) {
  const int c = (int)threadIdx.x;
  if (c >= 256) return;
  int L = 0, L1 = 0, L2 = 0, FI = 0, R = 0;
  int idx = 0;
  for (int l = 0; l < NL; ++l) {
    int fi = 0;
    for (int l1 = 0; l1 < LMAXP1; ++l1)
      for (int l2 = 0; l2 < LMAXP1; ++l2) {
        int dl = l1 - l2; if (dl < 0) dl = -dl;
        if (l < dl || l > l1 + l2) continue;
        for (int r = 0; r <= 2 * l; ++r) {
          if (idx == c) { L = l; L1 = l1; L2 = l2; FI = fi; R = r; }
          ++idx;
        }
        ++fi;
      }
  }
  Combo cb;
  cb.l = L; cb.l1 = L1; cb.l2 = L2; cb.fi = FI; cb.r = R;
  const int m = R - L;
  int K = 0;
  for (int pp = 0; pp <= 2 * L1; ++pp) {
    int m1 = pp - L1;
    int m2 = m - m1;
    if (m2 < -L2 || m2 > L2) continue;
    cb.p[K] = pp;
    cb.q[K] = m2 + L2;
    cb.c[K] = (float)cg_coef(L1, m1, L2, m2, L, m);
    ++K;
  }
  cb.K = K;
  for (int k = K; k < 8; ++k) { cb.p[k] = 0; cb.q[k] = 0; cb.c[k] = 0.0f; }
  cb.pad[0] = 0; cb.pad[1] = 0;
  combos[c] = cb;
}

// One block per (combo, batch): 64x64 complex tile via rank-2K f32 WMMA.
// A cols: [C_k*xr | C_k*xi], B_re rows: [yr ; -yi], B_im rows: [yi ; yr].
// B_re/B_im are pre-swizzled at staging so the WMMA loop is pure ds_load+wmma.
__global__ __launch_bounds__(256) void cg_prod_wmma(
    const float* __restrict__ x0, const float* __restrict__ x1,
    const float* __restrict__ x2, const float* __restrict__ x3,
    float* __restrict__ out, const Combo* __restrict__ combos) {
  __shared__ v2f xs[512];    // [k][i] = {C_k*xr, C_k*xi}, k padded to 8
  __shared__ v2f ysre[512];  // [k][j] = {yr, -yi}
  __shared__ v2f ysim[512];  // [k][j] = {yi,  yr}

  const int combo = (int)blockIdx.x >> 6;
  const int b = (int)blockIdx.x & 63;
  const Combo* __restrict__ cb = combos + combo;

  const int l  = cb->l;
  const int l1 = cb->l1;
  const int l2 = cb->l2;
  const int fi = cb->fi;
  const int r  = cb->r;
  const int K  = cb->K;

  const float* xp = (l1 == 0) ? x0 : (l1 == 1) ? x1 : (l1 == 2) ? x2 : x3;
  const float* yp = (l2 == 0) ? x0 : (l2 == 1) ? x1 : (l2 == 2) ? x2 : x3;
  const int n1 = 2 * l1 + 1, n2 = 2 * l2 + 1;

  // Stage scaled A-rows and pre-swizzled B-rows into LDS (zero-pad k >= K).
  for (int e = (int)threadIdx.x; e < 512; e += 256) {
    const int k = e >> 6, i = e & 63;
    v2f xv = {0.0f, 0.0f};
    v2f yre = {0.0f, 0.0f};
    v2f yim = {0.0f, 0.0f};
    if (k < K) {
      const float cgv = cb->c[k];
      const float* px = xp + ((size_t)((b * n1 + cb->p[k]) * TAU + i) << 1);
      xv.x = cgv * px[0];
      xv.y = cgv * px[1];
      const float* py = yp + ((size_t)((b * n2 + cb->q[k]) * TAU + i) << 1);
      const float yr = py[0], yi = py[1];
      yre.x = yr;  yre.y = -yi;
      yim.x = yi;  yim.y = yr;
    }
    xs[e] = xv;
    ysre[e] = yre;
    ysim[e] = yim;
  }
  __syncthreads();

  const int lane = (int)threadIdx.x & 31;
  const int w    = (int)threadIdx.x >> 5;
  const int half = lane >> 4;   // 0: K-slots {4kc,4kc+1}; 1: {4kc+2,4kc+3}
  const int lid  = lane & 15;
  const int nkc  = (K + 1) >> 1;  // 4 A-columns (= 2 complex pairs) per WMMA

  const int Tl = d_NFRAG[l] << 12;  // nfrag * 64*64
  const size_t base = (size_t)d_OUTBASE[l] +
      (((size_t)(b * (2 * l + 1) + r) * (size_t)Tl + ((size_t)fi << 12)) << 1);

  // 8 waves x 2 tiles = 16 tiles of 16x16 covering the 64x64 output.
  for (int t = 2 * w; t < 2 * w + 2; ++t) {
    const int ti = t >> 2, tj = t & 3;
    const int i = ti * 16 + lid;  // A row for this lane
    const int j = tj * 16 + lid;  // B col for this lane
    v8f cre = {};
    v8f cim = {};
    for (int kc = 0; kc < nkc; ++kc) {
      const int k = 2 * kc + half;             // complex pair index for this lane-half
      const v2f a   = xs[(k << 6) + i];        // A cols (2k, 2k+1) at row i
      const v2f bre = ysre[(k << 6) + j];      // B_re rows (2k, 2k+1) at col j
      const v2f bim = ysim[(k << 6) + j];      // B_im rows (2k, 2k+1) at col j
      cre = __builtin_amdgcn_wmma_f32_16x16x4_f32(
          false, a, false, bre, (short)0, cre, false, false);
      cim = __builtin_amdgcn_wmma_f32_16x16x4_f32(
          false, a, false, bim, (short)0, cim, false, false);
    }
    // D layout: VGPR v -> lanes 0-15: (M=v, N=lid); lanes 16-31: (M=v+8, N=lid)
#pragma unroll
    for (int v = 0; v < 8; ++v) {
      const int io = ti * 16 + v + (half << 3);
      const int jo = tj * 16 + lid;
      v2f o;
      o.x = cre[v];
      o.y = cim[v];
      *(v2f*)(out + base + (((size_t)(io * TAU + jo)) << 1)) = o;  // {re,im} 8B store
    }
  }
}

extern "C" void kernel_launch(void* const* d_in, const int* in_sizes, int n_in,
                              void* d_out, int out_size, void* d_ws, size_t ws_size,
                              hipStream_t stream) {
  (void)in_sizes; (void)n_in; (void)out_size; (void)ws_size;
  const float* x0 = (const float*)d_in[0];
  const float* x1 = (const float*)d_in[1];
  const float* x2 = (const float*)d_in[2];
  const float* x3 = (const float*)d_in[3];
  Combo* combos = (Combo*)d_ws;  // 256 * 128B = 32 KB

  cg_setup<<<1, 256, 0, stream>>>(combos);
  cg_prod_wmma<<<dim3(256 * BATCH), dim3(256), 0, stream>>>(
      x0, x1, x2, x3, (float*)d_out, combos);
}